// DHT_Layer_16527034155156
// MI455X (gfx1250) — compile-verified
//
#include <hip/hip_runtime.h>

typedef float v2f __attribute__((ext_vector_type(2)));
typedef float v8f __attribute__((ext_vector_type(8)));

#define HW   16384
#define CIN  128
#define CH   64
#define NA   180
#define NR   184
#define NP   (NA * NR)   /* 33120 */
#define BEPS 1e-5f

__device__ __forceinline__ v8f zero8() {
    v8f z = {0.f, 0.f, 0.f, 0.f, 0.f, 0.f, 0.f, 0.f};
    return z;
}

__device__ __forceinline__ v8f wmma_f32x4(v2f a, v2f b, v8f c) {
    // D = A(16x4 f32) * B(4x16 f32) + C(16x16 f32)
    return __builtin_amdgcn_wmma_f32_16x16x4_f32(false, a, false, b, (short)0, c, false, false);
}

__device__ __forceinline__ void wait_async0() {
#if __has_builtin(__builtin_amdgcn_s_wait_asynccnt)
    __builtin_amdgcn_s_wait_asynccnt(0);
#else
    asm volatile("s_wait_asynccnt 0x0" ::: "memory");
#endif
}

// pair-row swizzle: entry index for (pair-row pi, column c) in a [P][64] float2 tile.
// Odd pair-rows are rotated by 16 so the two half-waves of a b64 fragment read
// hit disjoint bank halves.
__device__ __forceinline__ int pswz(int pi, int c) {
    return pi * CH + ((c + ((pi & 1) << 4)) & 63);
}

// ---------------------------------------------------------------------------
// Kernel 1: fused 1x1 conv (Cin=128 -> 64) + bias + BN + ReLU.
// GEMM: M = pixels, N = out channel (64), K = Cin (128).
// Output channel-last: h1[b][p][c] so the DHT reads 64ch contiguously.
// ---------------------------------------------------------------------------
__global__ __launch_bounds__(256) void k_conv1x1(
    const float* __restrict__ x, const float* __restrict__ w,
    const float* __restrict__ bfirst,
    const float* __restrict__ g, const float* __restrict__ be,
    const float* __restrict__ mn, const float* __restrict__ vr,
    float* __restrict__ h1)
{
    __shared__ float2 wP[(CIN / 2) * CH];   // paired [cin/2][d] (swizzled), 32KB
    __shared__ float sc[CH], sh[CH];
    const int tid = threadIdx.x;
    const int b   = blockIdx.y;

    float* wPf = (float*)wP;
    for (int idx = tid; idx < CIN * CH; idx += 256) {
        int d = idx >> 7, c = idx & 127;          // w: [64][128] row-major, coalesced
        wPf[pswz(c >> 1, d) * 2 + (c & 1)] = w[idx];
    }
    if (tid < CH) {
        float inv = g[tid] * rsqrtf(vr[tid] + BEPS);
        sc[tid] = inv;
        sh[tid] = (bfirst[tid] - mn[tid]) * inv + be[tid];
    }
    __syncthreads();

    const int wv   = tid >> 5;
    const int lane = tid & 31;
    const int half = lane >> 4;
    const int l16  = lane & 15;
    const int pw   = blockIdx.x * 256 + wv * 32;   // 32 pixels per wave
    const float* xb = x + (size_t)b * CIN * HW;
    const v2f* wPv = (const v2f*)wP;

    v8f acc[2][4];
    for (int m = 0; m < 2; ++m)
        for (int n = 0; n < 4; ++n) acc[m][n] = zero8();

    for (int cb = 0; cb < CIN; cb += 4) {
        const int k0 = cb + half * 2;
        const int pi = k0 >> 1;                    // pair-row
        v2f afr[2];
        for (int m = 0; m < 2; ++m) {
            int p = pw + m * 16 + l16;
            afr[m].x = xb[(size_t)k0 * HW + p];
            afr[m].y = xb[(size_t)(k0 + 1) * HW + p];
        }
        for (int n = 0; n < 4; ++n) {
            v2f bfr = wPv[pswz(pi, n * 16 + l16)];   // one ds_load_b64
            for (int m = 0; m < 2; ++m)
                acc[m][n] = wmma_f32x4(afr[m], bfr, acc[m][n]);
        }
    }

    for (int m = 0; m < 2; ++m)
        for (int n = 0; n < 4; ++n) {
            const int d = n * 16 + l16;
            const float s = sc[d], t = sh[d];
            for (int i = 0; i < 8; ++i) {
                int p = pw + m * 16 + i + half * 8;
                float v = acc[m][n][i] * s + t;
                v = v > 0.f ? v : 0.f;
                h1[((size_t)b * HW + p) * CH + d] = v;
            }
        }
}

// ---------------------------------------------------------------------------
// Kernel 2: DHT. One block per (b, angle); LDS accumulator [184 rho][64 ch]
// (swizzled), ds_add_f32 atomics, then paired-channel coalesced writes:
// hough2[b][c/2][a][r] = float2{ch c, ch c+1}.
// ---------------------------------------------------------------------------
__global__ __launch_bounds__(256) void k_dht(
    const float* __restrict__ h1, float2* __restrict__ hough2)
{
    __shared__ float lacc[NR * CH];
    const int tid  = threadIdx.x;
    const int wv   = tid >> 5;
    const int lane = tid & 31;
    const int a    = blockIdx.x;
    const int b    = blockIdx.y;

    for (int i = tid; i < NR * CH; i += 256) lacc[i] = 0.f;
    __syncthreads();

    const float th = (float)a * 0.017453292519943295f;  // pi/180
    const float ct = cosf(th), st = sinf(th);
    const float* hb = h1 + (size_t)b * HW * CH;
    const int c0 = lane * 2;

    for (int p = wv; p < HW; p += 8) {
        float xf = (float)(p & 127) - 64.f;
        float yf = (float)(p >> 7) - 64.f;
        int r = (int)rintf(ct * xf + st * yf) + NR / 2;   // ties-to-even like jnp.round
        r = r < 0 ? 0 : (r > NR - 1 ? NR - 1 : r);
        const float2 v = *(const float2*)(hb + (size_t)p * CH + c0);
        atomicAdd(&lacc[r * CH + ((c0 + r) & 63)], v.x);
        atomicAdd(&lacc[r * CH + ((c0 + 1 + r) & 63)], v.y);
    }
    __syncthreads();

    for (int c2 = wv; c2 < CH / 2; c2 += 8)
        for (int r = lane; r < NR; r += 32) {
            float2 stv;
            stv.x = lacc[r * CH + ((2 * c2 + r) & 63)];
            stv.y = lacc[r * CH + ((2 * c2 + 1 + r) & 63)];
            hough2[(((size_t)b * (CH / 2) + c2) * NA + a) * NR + r] = stv;
        }
}

// ---------------------------------------------------------------------------
// Kernel 3: 3x3 conv (64->64) + BN + ReLU, implicit GEMM on WMMA f32 16x16x4.
// Input is channel-paired: in2[b][ci/2][p] = float2{ci, ci+1}  -> B-frag is one
// global_load_b64. Weights staged per-tap into LDS via async-to-LDS copies in
// paired layout -> A-frag is one ds_load_b64. Output paired for inner layers,
// flat [B][64][A][R] for the last layer.
// ---------------------------------------------------------------------------
__global__ __launch_bounds__(256) void k_conv3x3(
    const float2* __restrict__ in2, const float* __restrict__ w, // w: [64][64][3][3]
    const float* __restrict__ g, const float* __restrict__ be,
    const float* __restrict__ mn, const float* __restrict__ vr,
    float* __restrict__ out, int paired_out)
{
    __shared__ float2 wP[(CH / 2) * CH];   // paired [ci/2][co] (swizzled), 16KB
    __shared__ float sc[CH], sh[CH];
    const int tid = threadIdx.x;
    const int b   = blockIdx.y;

    if (tid < CH) {
        float inv = g[tid] * rsqrtf(vr[tid] + BEPS);
        sc[tid] = inv;
        sh[tid] = be[tid] - mn[tid] * inv;
    }

    const int wv   = tid >> 5;
    const int lane = tid & 31;
    const int half = lane >> 4;
    const int l16  = lane & 15;
    const int pw   = blockIdx.x * 256 + wv * 32;
    const float2* inb = in2 + (size_t)b * (CH / 2) * NP;
    const v2f* wPv = (const v2f*)wP;
    const unsigned wP_lds = (unsigned)(uintptr_t)&wP[0];   // LDS byte offset

    // Per-lane halo offsets/masks for the 9 taps of each N-tile.
    int offs[2][9];
    unsigned okm[2] = {0u, 0u};
    for (int nt = 0; nt < 2; ++nt) {
        int p = pw + nt * 16 + l16;
        bool pok = p < NP;
        int aa = p / NR, rr = p % NR;
        for (int t = 0; t < 9; ++t) {
            int a2 = aa + t / 3 - 1;
            int r2 = rr + t % 3 - 1;
            offs[nt][t] = a2 * NR + r2;
            bool ok = pok && a2 >= 0 && a2 < NA && r2 >= 0 && r2 < NR;
            if (ok) okm[nt] |= (1u << t);
        }
    }

    v8f acc[4][2];
    for (int m = 0; m < 4; ++m)
        for (int n = 0; n < 2; ++n) acc[m][n] = zero8();

    for (int tap = 0; tap < 9; ++tap) {
        __syncthreads();   // previous tap's LDS reads done
        // async stage this tap's 64x64 weight slice into LDS (paired+swizzled)
        for (int idx = tid; idx < CH * CH; idx += 256) {
            int co = idx >> 6, ci = idx & 63;
            unsigned dst = wP_lds + (unsigned)(pswz(ci >> 1, co) * 8 + (ci & 1) * 4);
            const float* src = w + co * 576 + ci * 9 + tap;
            asm volatile("global_load_async_to_lds_b32 %0, %1, off"
                         :: "v"(dst), "v"(src) : "memory");
        }
        wait_async0();
        __syncthreads();

        for (int cb = 0; cb < CH; cb += 4) {
            const int k0 = cb + half * 2;
            const int pi = k0 >> 1;                 // paired ci row
            v2f afr[4];
            for (int m = 0; m < 4; ++m)
                afr[m] = wPv[pswz(pi, m * 16 + l16)];       // ds_load_b64
            for (int nt = 0; nt < 2; ++nt) {
                v2f bfr = {0.f, 0.f};
                if ((okm[nt] >> tap) & 1u) {
                    const float2 t2 = inb[(size_t)pi * NP + offs[nt][tap]]; // global_load_b64
                    bfr.x = t2.x;
                    bfr.y = t2.y;
                }
                for (int m = 0; m < 4; ++m)
                    acc[m][nt] = wmma_f32x4(afr[m], bfr, acc[m][nt]);
            }
        }
    }

    float2* out2 = (float2*)out;
    for (int m = 0; m < 4; ++m)
        for (int nt = 0; nt < 2; ++nt) {
            int p = pw + nt * 16 + l16;
            if (p < NP) {
                if (paired_out) {
                    for (int i2 = 0; i2 < 4; ++i2) {
                        int co = m * 16 + half * 8 + i2 * 2;
                        float v0 = acc[m][nt][i2 * 2] * sc[co] + sh[co];
                        float v1 = acc[m][nt][i2 * 2 + 1] * sc[co + 1] + sh[co + 1];
                        float2 stv;
                        stv.x = v0 > 0.f ? v0 : 0.f;
                        stv.y = v1 > 0.f ? v1 : 0.f;
                        out2[((size_t)b * (CH / 2) + (co >> 1)) * NP + p] = stv;
                    }
                } else {
                    for (int i = 0; i < 8; ++i) {
                        int co = m * 16 + half * 8 + i;
                        float v = acc[m][nt][i] * sc[co] + sh[co];
                        v = v > 0.f ? v : 0.f;
                        out[((size_t)b * CH + co) * NP + p] = v;
                    }
                }
            }
        }
}

// ---------------------------------------------------------------------------
extern "C" void kernel_launch(void* const* d_in, const int* in_sizes, int n_in,
                              void* d_out, int out_size, void* d_ws, size_t ws_size,
                              hipStream_t stream)
{
    const float* x       = (const float*)d_in[0];
    const float* w_first = (const float*)d_in[1];
    const float* b_first = (const float*)d_in[2];
    const float* w_convs = (const float*)d_in[3];
    const float* bn_g    = (const float*)d_in[4];
    const float* bn_b    = (const float*)d_in[5];
    const float* bn_m    = (const float*)d_in[6];
    const float* bn_v    = (const float*)d_in[7];
    float* out = (float*)d_out;

    char* ws = (char*)d_ws;
    float*  h1    = (float*)ws;                                           // 8*16384*64 f32 = 32MB
    float2* hough = (float2*)(ws + (size_t)8 * HW * CH * 4);              // paired, 64.7MB
    float*  buf1  = (float*)(ws + (size_t)8 * HW * CH * 4 +
                                  (size_t)8 * CH * NP * 4);               // paired, 64.7MB

    const int B = 8;
    k_conv1x1<<<dim3(HW / 256, B), 256, 0, stream>>>(
        x, w_first, b_first, bn_g, bn_b, bn_m, bn_v, h1);
    k_dht<<<dim3(NA, B), 256, 0, stream>>>(h1, hough);

    const int pblk = (NP + 255) / 256;   // 130
    const size_t WSTEP = (size_t)CH * CH * 9;
    // chain: hough -> buf1 -> d_out(paired scratch) -> buf1 -> d_out(flat final)
    k_conv3x3<<<dim3(pblk, B), 256, 0, stream>>>(
        hough,          w_convs + 0 * WSTEP, bn_g + 64,  bn_b + 64,  bn_m + 64,  bn_v + 64,  buf1, 1);
    k_conv3x3<<<dim3(pblk, B), 256, 0, stream>>>(
        (float2*)buf1,  w_convs + 1 * WSTEP, bn_g + 128, bn_b + 128, bn_m + 128, bn_v + 128, out,  1);
    k_conv3x3<<<dim3(pblk, B), 256, 0, stream>>>(
        (float2*)out,   w_convs + 2 * WSTEP, bn_g + 192, bn_b + 192, bn_m + 192, bn_v + 192, buf1, 1);
    k_conv3x3<<<dim3(pblk, B), 256, 0, stream>>>(
        (float2*)buf1,  w_convs + 3 * WSTEP, bn_g + 256, bn_b + 256, bn_m + 256, bn_v + 256, out,  0);
}